// GNNActor_54408645705762
// MI455X (gfx1250) — compile-verified
//
#include <hip/hip_runtime.h>
#include <hip/hip_bf16.h>
#include <math.h>

#define STATE 16
#define EDIM  4
#define HID   64
#define OUTC  16

typedef __attribute__((ext_vector_type(2))) float v2f;
typedef __attribute__((ext_vector_type(8))) float v8f;

// V_WMMA_F32_16X16X4_F32: D(16x16 f32) = A(16x4 f32) * B(4x16 f32) + C
static __device__ __forceinline__ v8f wmma4(v2f a, v2f b, v8f c) {
  return __builtin_amdgcn_wmma_f32_16x16x4_f32(false, a, false, b, (short)0, c,
                                               false, false);
}

// ---------------- init: slot map = -1, zero accumulators ----------------
__global__ void k_init(int* slot, float* msg_sum, float* cnt, int N, int M) {
  int i = blockIdx.x * blockDim.x + threadIdx.x;
  if (i < N) slot[i] = -1;
  if (i < M * OUTC) msg_sum[i] = 0.0f;
  if (i < M) cnt[i] = 0.0f;
}

__global__ void k_scatter(int* slot, const int* __restrict__ mask_idx, int M) {
  int i = blockIdx.x * blockDim.x + threadIdx.x;
  if (i < M) slot[mask_idx[i]] = i;
}

// ---------------- edge kernel: 1 wave = 16 edges, WMMA f32 ----------------
__global__ void __launch_bounds__(32)
k_edges(const float* __restrict__ x, const float* __restrict__ edge_attr,
        const float* __restrict__ w_e1, const float* __restrict__ b_e1,
        const float* __restrict__ w_e2, const float* __restrict__ b_e2,
        const int* __restrict__ edge_index, const int* __restrict__ slot,
        float* msg_sum, float* cnt, int E)
{
  __shared__ __align__(16) float s_attr[16][EDIM];
  __shared__ __align__(16) float s_x[16][STATE];   // x[src[e]]
  __shared__ __align__(16) float s_h[16][16];      // relu(attr@w_e1+b_e1)
  __shared__ int s_slot[16];
  __shared__ int s_any;

  const int lane = threadIdx.x;
  const int hi = lane >> 4;        // 0: K/row group low, 1: high
  const int ln = lane & 15;
  const long e0 = (long)blockIdx.x * 16;

  if (lane == 31) s_any = 0;
  __syncthreads();

  if (lane < 16) {
    long e = e0 + lane;
    int s = -1;
    if (e < (long)E) {
      int dst = edge_index[(long)E + e];   // edge_index[1][e]
      s = slot[dst];
    }
    s_slot[lane] = s;
    if (s >= 0) {
      int src = edge_index[e];             // edge_index[0][e]
      const float4 ea = *(const float4*)(edge_attr + e * EDIM);
      s_attr[lane][0] = ea.x; s_attr[lane][1] = ea.y;
      s_attr[lane][2] = ea.z; s_attr[lane][3] = ea.w;
      const float4* xr = (const float4*)(x + (long)src * STATE);
      #pragma unroll
      for (int q = 0; q < 4; ++q) ((float4*)s_x[lane])[q] = xr[q];
      s_any = 1;
      atomicAdd(&cnt[s], 1.0f);            // full in-degree of mask node
    } else {
      #pragma unroll
      for (int q = 0; q < EDIM; ++q) s_attr[lane][q] = 0.0f;
      #pragma unroll
      for (int q = 0; q < 4; ++q) ((float4*)s_x[lane])[q] = make_float4(0,0,0,0);
    }
  }
  __syncthreads();
  if (!s_any) return;   // uniform

  // H = relu(attr @ w_e1 + b_e1)  -- single K=4 f32 WMMA
  {
    v2f a, b; v8f c;
    a.x = s_attr[ln][2*hi + 0];                 // A 16x4 layout
    a.y = s_attr[ln][2*hi + 1];
    b.x = w_e1[(2*hi + 0) * 16 + ln];           // B 4x16 layout
    b.y = w_e1[(2*hi + 1) * 16 + ln];
    float bb = b_e1[ln];
    #pragma unroll
    for (int r = 0; r < 8; ++r) c[r] = bb;
    c = wmma4(a, b, c);
    #pragma unroll
    for (int r = 0; r < 8; ++r) s_h[r + 8*hi][ln] = fmaxf(c[r], 0.0f);
  }
  __syncthreads();

  // W tile t = (H @ w_e2 + b_e2)[:, t*16 .. t*16+15] == W[e, i=t, o=0..15]
  // contract immediately: msg[e,o] += x_src[e, t] * Wtile[e, o]
  float macc[8];
  #pragma unroll
  for (int r = 0; r < 8; ++r) macc[r] = 0.0f;

  #pragma unroll 1
  for (int t = 0; t < 16; ++t) {
    v8f cw;
    float bb = b_e2[t * 16 + ln];
    #pragma unroll
    for (int r = 0; r < 8; ++r) cw[r] = bb;
    #pragma unroll
    for (int kc = 0; kc < 4; ++kc) {            // K=16 in 4 WMMA steps
      v2f aw, bw;
      aw.x = s_h[ln][4*kc + 2*hi + 0];
      aw.y = s_h[ln][4*kc + 2*hi + 1];
      bw.x = w_e2[(4*kc + 2*hi + 0) * 256 + t * 16 + ln];
      bw.y = w_e2[(4*kc + 2*hi + 1) * 256 + t * 16 + ln];
      cw = wmma4(aw, bw, cw);
    }
    #pragma unroll
    for (int r = 0; r < 8; ++r)
      macc[r] += s_x[r + 8*hi][t] * cw[r];      // D layout: row r+8*hi, col ln
  }

  #pragma unroll
  for (int r = 0; r < 8; ++r) {
    int er = r + 8*hi;
    int s = s_slot[er];
    if (s >= 0) atomicAdd(&msg_sum[s * OUTC + ln], macc[r]);
  }
}

// ---------------- y = msg/max(cnt,1) + x@root + bias (mask rows only) ----
__global__ void k_y(const float* __restrict__ x, const float* __restrict__ root,
                    const float* __restrict__ bias, const int* __restrict__ mask_idx,
                    const float* __restrict__ msg_sum, const float* __restrict__ cnt,
                    float* yv, int M)
{
  int i = blockIdx.x * blockDim.x + threadIdx.x;
  if (i >= M * OUTC) return;
  int m = i >> 4, o = i & 15;
  int n = mask_idx[m];
  float acc = msg_sum[i] / fmaxf(cnt[m], 1.0f) + bias[o];
  const float* xr = x + (long)n * STATE;
  #pragma unroll
  for (int k = 0; k < STATE; ++k) acc = fmaf(xr[k], root[k * OUTC + o], acc);
  yv[i] = acc;
}

// ---------------- MLP head: 1 wave = 16 rows, WMMA f32 ----------------
__global__ void __launch_bounds__(32)
k_head(const float* __restrict__ yv,
       const float* __restrict__ fc1_w, const float* __restrict__ fc1_b,
       const float* __restrict__ fc2_w, const float* __restrict__ fc2_b,
       const float* __restrict__ fc3_w, const float* __restrict__ fc3_b,
       float* __restrict__ out, int M)
{
  __shared__ __align__(16) float s_y[16][16];
  __shared__ __align__(16) float s_h1[16][HID];
  __shared__ __align__(16) float s_h2[16][HID];
  const int lane = threadIdx.x, hi = lane >> 4, ln = lane & 15;
  const int row0 = blockIdx.x * 16;

  if (lane < 16) {
    int row = row0 + lane;
    if (row < M) {
      #pragma unroll
      for (int q = 0; q < 4; ++q)
        ((float4*)s_y[lane])[q] = ((const float4*)(yv + (long)row * 16))[q];
    } else {
      #pragma unroll
      for (int q = 0; q < 16; ++q) s_y[lane][q] = 0.0f;
    }
  }
  __syncthreads();

  // h1 = relu(y @ fc1 + b1): [16x16]@[16x64], 4 col tiles x 4 K-chunks
  #pragma unroll 1
  for (int t = 0; t < 4; ++t) {
    v8f c; float bb = fc1_b[t*16 + ln];
    #pragma unroll
    for (int r = 0; r < 8; ++r) c[r] = bb;
    #pragma unroll
    for (int kc = 0; kc < 4; ++kc) {
      v2f a, b;
      a.x = s_y[ln][4*kc + 2*hi + 0];
      a.y = s_y[ln][4*kc + 2*hi + 1];
      b.x = fc1_w[(4*kc + 2*hi + 0) * HID + t*16 + ln];
      b.y = fc1_w[(4*kc + 2*hi + 1) * HID + t*16 + ln];
      c = wmma4(a, b, c);
    }
    #pragma unroll
    for (int r = 0; r < 8; ++r) s_h1[r + 8*hi][t*16 + ln] = fmaxf(c[r], 0.0f);
  }
  __syncthreads();

  // h2 = relu(h1 @ fc2 + b2): [16x64]@[64x64], 4 col tiles x 16 K-chunks
  #pragma unroll 1
  for (int t = 0; t < 4; ++t) {
    v8f c; float bb = fc2_b[t*16 + ln];
    #pragma unroll
    for (int r = 0; r < 8; ++r) c[r] = bb;
    #pragma unroll
    for (int kc = 0; kc < 16; ++kc) {
      v2f a, b;
      a.x = s_h1[ln][4*kc + 2*hi + 0];
      a.y = s_h1[ln][4*kc + 2*hi + 1];
      b.x = fc2_w[(4*kc + 2*hi + 0) * HID + t*16 + ln];
      b.y = fc2_w[(4*kc + 2*hi + 1) * HID + t*16 + ln];
      c = wmma4(a, b, c);
    }
    #pragma unroll
    for (int r = 0; r < 8; ++r) s_h2[r + 8*hi][t*16 + ln] = fmaxf(c[r], 0.0f);
  }
  __syncthreads();

  if (lane < 16) {
    int row = row0 + lane;
    if (row < M) {
      float o0 = fc3_b[0], o1 = fc3_b[1];
      #pragma unroll
      for (int k = 0; k < HID; ++k) {
        float v = s_h2[lane][k];
        o0 = fmaf(v, fc3_w[k*2 + 0], o0);
        o1 = fmaf(v, fc3_w[k*2 + 1], o1);
      }
      o0 = tanhf(o0); o1 = tanhf(o1);
      out[row] = atan2f(o0, o1) * 0.31830988618379067154f;  // /pi
    }
  }
}

// ---------------- host-side launch ----------------
extern "C" void kernel_launch(void* const* d_in, const int* in_sizes, int n_in,
                              void* d_out, int out_size, void* d_ws, size_t ws_size,
                              hipStream_t stream)
{
  const float* x         = (const float*)d_in[0];
  const float* edge_attr = (const float*)d_in[1];
  const float* w_e1      = (const float*)d_in[2];
  const float* b_e1      = (const float*)d_in[3];
  const float* w_e2      = (const float*)d_in[4];
  const float* b_e2      = (const float*)d_in[5];
  const float* root      = (const float*)d_in[6];
  const float* bias      = (const float*)d_in[7];
  const float* fc1_w     = (const float*)d_in[8];
  const float* fc1_b     = (const float*)d_in[9];
  const float* fc2_w     = (const float*)d_in[10];
  const float* fc2_b     = (const float*)d_in[11];
  const float* fc3_w     = (const float*)d_in[12];
  const float* fc3_b     = (const float*)d_in[13];
  const int* edge_index  = (const int*)d_in[14];
  const int* mask_idx    = (const int*)d_in[15];

  const int N = in_sizes[0] / STATE;
  const int E = in_sizes[1] / EDIM;
  const int M = in_sizes[15];

  char* w = (char*)d_ws;
  int*   slot    = (int*)w;    w += (size_t)N * sizeof(int);
  float* msg_sum = (float*)w;  w += (size_t)M * OUTC * sizeof(float);
  float* cnt     = (float*)w;  w += (size_t)M * sizeof(float);
  float* yv      = (float*)w;  w += (size_t)M * OUTC * sizeof(float);
  (void)ws_size; (void)n_in; (void)out_size;

  int initN = (N > M * OUTC) ? N : M * OUTC;
  k_init<<<(initN + 255) / 256, 256, 0, stream>>>(slot, msg_sum, cnt, N, M);
  k_scatter<<<(M + 255) / 256, 256, 0, stream>>>(slot, mask_idx, M);
  k_edges<<<(E + 15) / 16, 32, 0, stream>>>(x, edge_attr, w_e1, b_e1, w_e2, b_e2,
                                            edge_index, slot, msg_sum, cnt, E);
  k_y<<<(M * OUTC + 255) / 256, 256, 0, stream>>>(x, root, bias, mask_idx,
                                                  msg_sum, cnt, yv, M);
  k_head<<<(M + 15) / 16, 32, 0, stream>>>(yv, fc1_w, fc1_b, fc2_w, fc2_b,
                                           fc3_w, fc3_b, (float*)d_out, M);
}